// GAT_57432302682311
// MI455X (gfx1250) — compile-verified
//
#include <hip/hip_runtime.h>
#include <hip/hip_bf16.h>

// ---------------------------------------------------------------------------
// 3-layer GAT for MI455X (gfx1250, wave32).
// GEMMs use V_WMMA_F32_16X16X4_F32 (exact fp32 on the matrix pipe).
// Edge softmax / aggregation use native f32 global atomics.
// ---------------------------------------------------------------------------

typedef float v2f __attribute__((ext_vector_type(2)));
typedef float v4f __attribute__((ext_vector_type(4)));
typedef float v8f __attribute__((ext_vector_type(8)));

#define N_NODES 50000
#define N_EDGES 500000
#define SLOPE   0.2f

// ---------------------------------------------------------------------------
// C[M,N] = A[M,K] * B[K,N], row-major, fp32 WMMA. One wave per 16x32 C tile
// (two accumulators sharing the A fragment -> 2 WMMAs per A load).
// grid = (M/16, N/32), block = 32. K and N are compile-time so all B-side
// addresses inside the unrolled body fold into 24-bit instruction offsets.
// A 16x4 layout (ISA 7.12.2): lanes 0-15 -> rows 0-15 @ K={0,1};
//                             lanes 16-31 -> rows 0-15 @ K={2,3}.
// B 4x16 mirrored; C: VGPR r -> row hi*8+r, col = lane&15.
// ---------------------------------------------------------------------------
template <int K, int N>
__global__ __launch_bounds__(32) void gat_gemm_wmma_f32(
    const float* __restrict__ A, const float* __restrict__ B,
    float* __restrict__ C) {
  const int tile_m = blockIdx.x;
  const int tile_n = blockIdx.y;        // 32-wide N tile
  const int lane   = threadIdx.x;
  const int hi     = lane >> 4;         // 0: K base 0, 1: K base 2
  const int l15    = lane & 15;

  const float* aptr = A + (size_t)(tile_m * 16 + l15) * K + hi * 2;
  const float* bptr = B + (size_t)(hi * 2) * N + tile_n * 32 + l15;

  v8f acc0 = {};
  v8f acc1 = {};
#pragma unroll 8
  for (int k = 0; k < K; k += 4) {
    v2f a, b0, b1;
    a.x  = aptr[k];                     // contiguous pair -> global_load_b64
    a.y  = aptr[k + 1];
    b0.x = bptr[(size_t)k * N];         // k*N folds to an immediate offset
    b0.y = bptr[(size_t)(k + 1) * N];
    b1.x = bptr[(size_t)k * N + 16];
    b1.y = bptr[(size_t)(k + 1) * N + 16];
    acc0 = __builtin_amdgcn_wmma_f32_16x16x4_f32(
        false, a, false, b0, (short)0, acc0, false, false);
    acc1 = __builtin_amdgcn_wmma_f32_16x16x4_f32(
        false, a, false, b1, (short)0, acc1, false, false);
  }

  float* cptr = C + (size_t)(tile_m * 16 + hi * 8) * N + tile_n * 32 + l15;
#pragma unroll
  for (int r = 0; r < 8; ++r) {
    cptr[(size_t)r * N]      = acc0[r];
    cptr[(size_t)r * N + 16] = acc1[r];
  }
}

// ---------------------------------------------------------------------------
// el[n,h] = sum_d ft[n,h,d]*al[h,d]; er likewise. One wave per (node,head).
// D fixed at 64 -> each lane reads 2 floats, d = lane and lane+32, shfl-xor reduce.
// ---------------------------------------------------------------------------
__global__ __launch_bounds__(256) void gat_attn_lr(
    const float* __restrict__ ft, const float* __restrict__ al,
    const float* __restrict__ ar, float* __restrict__ el,
    float* __restrict__ er, int items, int H) {
  int wave = (blockIdx.x * 256 + threadIdx.x) >> 5;
  int lane = threadIdx.x & 31;
  if (wave >= items) return;
  int h = wave % H;
  const float* f   = ft + (size_t)wave * 64;
  const float* alh = al + h * 64;
  const float* arh = ar + h * 64;
  float vl = f[lane] * alh[lane] + f[lane + 32] * alh[lane + 32];
  float vr = f[lane] * arh[lane] + f[lane + 32] * arh[lane + 32];
#pragma unroll
  for (int m = 16; m; m >>= 1) {
    vl += __shfl_xor(vl, m, 32);
    vr += __shfl_xor(vr, m, 32);
  }
  if (lane == 0) { el[wave] = vl; er[wave] = vr; }
}

// Correct float atomic-max via ordered integer encodings (native int atomics).
__device__ __forceinline__ void atomicMaxF(float* addr, float v) {
  if (v >= 0.0f)
    atomicMax((int*)addr, __float_as_int(v));
  else
    atomicMin((unsigned int*)addr, (unsigned int)__float_as_int(v));
}

// emax <- -FLT_MAX, denom <- 0, out <- bias broadcast
__global__ __launch_bounds__(256) void gat_init_layer(
    float* __restrict__ emax, float* __restrict__ denom,
    float* __restrict__ out, const float* __restrict__ bias,
    int NH, int NF, int F) {
  int t = blockIdx.x * 256 + threadIdx.x;
  if (t < NH) { emax[t] = -3.402823466e38f; denom[t] = 0.0f; }
  if (t < NF) { out[t] = bias[t % F]; }
}

// e = leaky_relu(el[src]+er[dst]); segment max over dst.
__global__ __launch_bounds__(256) void gat_edge_scores(
    const int* __restrict__ src, const int* __restrict__ dst,
    const float* __restrict__ el, const float* __restrict__ er,
    float* __restrict__ ebuf, float* __restrict__ emax, int E, int H) {
  int t = blockIdx.x * 256 + threadIdx.x;
  if (t >= E * H) return;
  int e = t / H, h = t - e * H;
  int s = src[e], d = dst[e];
  float x  = el[s * H + h] + er[d * H + h];
  float sc = x > 0.0f ? x : SLOPE * x;
  ebuf[t] = sc;
  atomicMaxF(&emax[d * H + h], sc);
}

// ee = exp(e - emax[dst]); segment sum over dst.
__global__ __launch_bounds__(256) void gat_edge_exp(
    const int* __restrict__ dst, const float* __restrict__ emax,
    float* __restrict__ ebuf, float* __restrict__ denom, int E, int H) {
  int t = blockIdx.x * 256 + threadIdx.x;
  if (t >= E * H) return;
  int e = t / H, h = t - e * H;
  int d = dst[e];
  float ee = __expf(ebuf[t] - emax[d * H + h]);
  ebuf[t] = ee;
  atomicAdd(&denom[d * H + h], ee);
}

// out[dst] += ft[src] * (ee/denom[dst]); one thread per (edge, 4 features).
// ft gathers become global_load_b128; atomics remain native f32 adds.
__global__ __launch_bounds__(256) void gat_edge_aggregate(
    const int* __restrict__ src, const int* __restrict__ dst,
    const float* __restrict__ ft, const float* __restrict__ ebuf,
    const float* __restrict__ denom, float* __restrict__ out,
    int E, int H, int D) {
  int F4 = (H * D) >> 2;                     // feature groups of 4
  int t = blockIdx.x * 256 + threadIdx.x;    // E*F4 <= 32M < 2^31
  if (t >= E * F4) return;
  int e  = t / F4, g = t - e * F4;
  int f  = g << 2;                            // first feature of group
  int h  = f / D;                             // D multiple of 4: no straddle
  int s  = src[e], d = dst[e];
  float alpha = ebuf[e * H + h] / denom[d * H + h];
  const v4f* fsrc = (const v4f*)(ft + (size_t)s * (H * D) + f);
  v4f v = *fsrc;
  float* o = out + (size_t)d * (H * D) + f;
  atomicAdd(o + 0, v.x * alpha);
  atomicAdd(o + 1, v.y * alpha);
  atomicAdd(o + 2, v.z * alpha);
  atomicAdd(o + 3, v.w * alpha);
}

extern "C" void kernel_launch(void* const* d_in, const int* in_sizes, int n_in,
                              void* d_out, int out_size, void* d_ws, size_t ws_size,
                              hipStream_t stream) {
  const float* feats = (const float*)d_in[0];
  const int*   src   = (const int*)d_in[1];
  const int*   dst   = (const int*)d_in[2];
  const float* W1  = (const float*)d_in[3];
  const float* al1 = (const float*)d_in[4];
  const float* ar1 = (const float*)d_in[5];
  const float* b1  = (const float*)d_in[6];
  const float* W2  = (const float*)d_in[7];
  const float* al2 = (const float*)d_in[8];
  const float* ar2 = (const float*)d_in[9];
  const float* b2  = (const float*)d_in[10];
  const float* W3  = (const float*)d_in[11];
  const float* al3 = (const float*)d_in[12];
  const float* ar3 = (const float*)d_in[13];
  const float* b3  = (const float*)d_in[14];

  const int N = N_NODES, E = N_EDGES;

  // Workspace layout (floats). Total ~28.4M floats = ~114 MB.
  float* ws    = (float*)d_ws;
  float* ftb   = ws;                       // 50000*256 = 12,800,000
  float* hbuf  = ftb  + (size_t)N * 256;   // 12,800,000
  float* el    = hbuf + (size_t)N * 256;   // 200,000
  float* er    = el   + (size_t)N * 4;     // 200,000
  float* emax  = er   + (size_t)N * 4;     // 200,000
  float* denom = emax + (size_t)N * 4;     // 200,000
  float* ebuf  = denom+ (size_t)N * 4;     // 2,000,000
  (void)ws_size; (void)in_sizes; (void)n_in; (void)out_size;

  auto layer = [&](const float* hin, const float* W, const float* al,
                   const float* ar, const float* bias, float* ft_l,
                   float* hout, int K, int H, int D) {
    const int F   = H * D;
    const int NH  = N * H;
    const int NF  = N * F;
    const int EH  = E * H;
    const int EF4 = E * (F >> 2);

    dim3 gg(N / 16, F / 32);
    if (K == 128)
      gat_gemm_wmma_f32<128, 256><<<gg, 32, 0, stream>>>(hin, W, ft_l);
    else if (F == 256)
      gat_gemm_wmma_f32<256, 256><<<gg, 32, 0, stream>>>(hin, W, ft_l);
    else
      gat_gemm_wmma_f32<256, 64><<<gg, 32, 0, stream>>>(hin, W, ft_l);
    gat_attn_lr<<<(NH + 7) / 8, 256, 0, stream>>>(ft_l, al, ar, el, er, NH, H);
    // hin is fully consumed by the GEMM above; safe to overwrite hout==hin.
    gat_init_layer<<<(NF + 255) / 256, 256, 0, stream>>>(emax, denom, hout, bias, NH, NF, F);
    gat_edge_scores<<<(EH + 255) / 256, 256, 0, stream>>>(src, dst, el, er, ebuf, emax, E, H);
    gat_edge_exp<<<(EH + 255) / 256, 256, 0, stream>>>(dst, emax, ebuf, denom, E, H);
    gat_edge_aggregate<<<(EF4 + 255) / 256, 256, 0, stream>>>(src, dst, ft_l, ebuf, denom, hout, E, H, D);
  };

  // Layer 1: 128 -> 4x64
  layer(feats, W1, al1, ar1, b1, ftb, hbuf, 128, 4, 64);
  // Layer 2: 256 -> 4x64 (in-place chain on hbuf is stream-safe)
  layer(hbuf, W2, al2, ar2, b2, ftb, hbuf, 256, 4, 64);
  // Layer 3: 256 -> 1x64, result directly to d_out
  layer(hbuf, W3, al3, ar3, b3, ftb, (float*)d_out, 256, 1, 64);
}